// Geodesic_Loss_19765439496953
// MI455X (gfx1250) — compile-verified
//
#include <hip/hip_runtime.h>

typedef __attribute__((ext_vector_type(2))) float v2f;
typedef __attribute__((ext_vector_type(8))) float v8f;

// SWAPX16 lane exchange: group-of-32 swizzle, xor_mask=0x10, and_mask=0x1f.
#define SWIZZLE_SWAPX16 0x401F

// Sum a per-lane f32 value across all 32 lanes of a wave using
// V_WMMA_F32_16X16X4_F32 (A = partials, B = ones). Result in every lane.
// Requires EXEC == all ones (guaranteed: called on uniform control paths).
__device__ __forceinline__ float wave_sum_wmma(float p) {
  v2f a; a[0] = p;    a[1] = 0.0f;   // A: 16x4 f32, lanes 0-15 -> K=0, lanes 16-31 -> K=2
  v2f b; b[0] = 1.0f; b[1] = 1.0f;   // B: 4x16 all-ones
  v8f c = {};
  // 8 args: (neg_a, A, neg_b, B, c_mod, C, reuse_a, reuse_b)
  c = __builtin_amdgcn_wmma_f32_16x16x4_f32(
      /*neg_a=*/false, a, /*neg_b=*/false, b,
      /*c_mod=*/(short)0, c, /*reuse_a=*/false, /*reuse_b=*/false);
  // D[m][n] = p_m + p_{m+16}. Lane holds rows {0..7} (+8 if lane>=16) in c[0..7].
  float s = ((c[0] + c[1]) + (c[2] + c[3])) + ((c[4] + c[5]) + (c[6] + c[7]));
  // lanes 0-15 hold sum of lanes {0..7,16..23}; lanes 16-31 the complement.
  // One ds_swizzle (SWAPX16) completes the 32-lane sum in every lane.
  int sw = __builtin_amdgcn_ds_swizzle(__builtin_bit_cast(int, s), SWIZZLE_SWAPX16);
  s += __builtin_bit_cast(float, sw);
  return s;
}

__device__ __forceinline__ float theta_of(const float* __restrict__ p,
                                          const float* __restrict__ g) {
  float dot = 0.0f;
#pragma unroll
  for (int k = 0; k < 9; ++k) dot = __builtin_fmaf(p[k], g[k], dot);
  float cosv = (dot - 1.0f) * 0.5f;
  cosv = fminf(1.0f, fmaxf(-1.0f, cosv));
  return acosf(cosv);
}

__global__ void __launch_bounds__(256)
geodesic_partial_kernel(const float* __restrict__ pred,
                        const float* __restrict__ gt,
                        float* __restrict__ partials,
                        int nBatch) {
  const int tid    = blockIdx.x * blockDim.x + threadIdx.x;
  const int stride = gridDim.x * blockDim.x;

  float acc = 0.0f;
  // Uniform (scalar) trip count: no per-iteration exec masking in the hot loop.
  const int iters = nBatch / stride;
  for (int j = 0; j < iters; ++j) {
    const size_t i = (size_t)tid + (size_t)j * stride;
    acc += theta_of(pred + i * 9, gt + i * 9);
  }
  // Tail (empty when stride divides nBatch, as with B = 2^21).
  {
    const int i = tid + iters * stride;
    if (i < nBatch) acc += theta_of(pred + (size_t)i * 9, gt + (size_t)i * 9);
  }

  // Wave reduction via WMMA, then deterministic LDS reduction across 8 waves.
  const float wsum = wave_sum_wmma(acc);

  __shared__ float lds[8];
  const int lane = threadIdx.x & 31;
  const int wave = threadIdx.x >> 5;
  if (lane == 0) lds[wave] = wsum;
  __syncthreads();
  if (threadIdx.x == 0) {
    float t = 0.0f;
#pragma unroll
    for (int w = 0; w < 8; ++w) t += lds[w];
    partials[blockIdx.x] = t;
  }
}

__global__ void __launch_bounds__(1024)
geodesic_final_kernel(const float* __restrict__ partials,
                      float* __restrict__ out,
                      int nPartials, float invB) {
  // Single block of 1024 threads (32 waves).
  const int lane = threadIdx.x & 31;
  const int wave = threadIdx.x >> 5;

  float v = (threadIdx.x < nPartials) ? partials[threadIdx.x] : 0.0f;
  const float wsum = wave_sum_wmma(v);

  __shared__ float lds[32];
  if (lane == 0) lds[wave] = wsum;
  __syncthreads();

  if (wave == 0) {
    float t = lds[lane];                 // 32 waves -> 32 values, one per lane
    const float total = wave_sum_wmma(t);
    if (lane == 0) out[0] = total * invB;
  }
}

extern "C" void kernel_launch(void* const* d_in, const int* in_sizes, int n_in,
                              void* d_out, int out_size, void* d_ws, size_t ws_size,
                              hipStream_t stream) {
  const float* pred = (const float*)d_in[0];
  const float* gt   = (const float*)d_in[1];
  float* out        = (float*)d_out;
  float* partials   = (float*)d_ws;

  const int nBatch  = in_sizes[0] / 9;      // B = 2,097,152
  const int threads = 256;
  const int blocks  = 1024;                 // 262,144 threads -> 8 batches/thread

  geodesic_partial_kernel<<<blocks, threads, 0, stream>>>(pred, gt, partials, nBatch);
  geodesic_final_kernel<<<1, 1024, 0, stream>>>(partials, out, blocks,
                                                1.0f / (float)nBatch);
}